// DeformConvBlock_60653528154323
// MI455X (gfx1250) — compile-verified
//
#include <hip/hip_runtime.h>
#include <hip/hip_bf16.h>

typedef float v2f __attribute__((ext_vector_type(2)));
typedef float v8f __attribute__((ext_vector_type(8)));

#define B_   4
#define C_   64
#define H_   256
#define W_   256
#define HW_  65536
#define OUT0_ELEMS (B_ * C_ * HW_)        // 16777216
#define OFF_ELEMS  (B_ * 18 * HW_)        // 4718592
#define K1_  576                           // 64 channels * 9 taps (im2col K)

// 8-byte load from a 4-byte-aligned address (gfx12 unaligned access ->
// global_load_b64; worst case the compiler splits it, which is no worse).
__device__ __forceinline__ v2f load2(const float* p) {
    v2f r;
    __builtin_memcpy(&r, p, sizeof(float) * 2);
    return r;
}

// im2col element for the offset conv: q indexes (c, ky, kx), zero padding.
__device__ __forceinline__ float im2col_load(const float* __restrict__ bp,
                                             int q, int y, int x) {
    int c  = q / 9;
    int r  = q - c * 9;
    int ky = r / 3;
    int kx = r - ky * 3;
    int yy = y + ky - 1, xx = x + kx - 1;
    int yc = min(max(yy, 0), H_ - 1);
    int xc = min(max(xx, 0), W_ - 1);
    float t = bp[(size_t)c * HW_ + yc * W_ + xc];
    bool v = (yy >= 0) && (yy < H_) && (xx >= 0) && (xx < W_);
    return v ? t : 0.f;
}

// ---------------------------------------------------------------------------
// Stage 1: 3x3 conv -> 18 offset channels, as im2col GEMM on WMMA f32.
// Wave tile: M = 16 pixels, N = 32 (channels 0..17 valid, rest masked),
// K = 576 stepped by 4. Weight panel (576x18, f32-pair packed) in LDS.
// ---------------------------------------------------------------------------
__global__ void __launch_bounds__(256) offset_conv_wmma_kernel(
    const float* __restrict__ blur, const float* __restrict__ w,
    const float* __restrict__ bias, float* __restrict__ offs)
{
    __shared__ v2f wp1[(K1_ / 2) * 18 + 16];   // [(q/2)*18 + o] = {w[o][q], w[o][q+1]}

    const int tid  = threadIdx.x;
    const int lane = tid & 31;
    const int wave = tid >> 5;
    const int m    = lane & 15;       // pixel (A/M) and out-channel (B/N) index
    const int half = lane >> 4;       // K sub-block select

    for (int i = tid; i < (K1_ / 2) * 18; i += 256) {
        int q2 = i / 18, o = i - q2 * 18;
        const float* src = w + o * K1_ + 2 * q2;
        v2f pv; pv.x = src[0]; pv.y = src[1];
        wp1[i] = pv;
    }
    __syncthreads();

    int tileIdx = blockIdx.x * 8 + wave;      // 16384 tiles exact cover
    int b   = tileIdx >> 12;
    int rem = tileIdx & 4095;
    int y   = rem >> 4;
    int xb  = (rem & 15) << 4;
    int x   = xb + m;

    const float* bp = blur + (size_t)b * C_ * HW_;
    const float gmask = (m < 2) ? 1.f : 0.f;   // N-tile1: only o=16,17 real

    v8f acc[2];
    acc[0] = (v8f){}; acc[1] = (v8f){};

#pragma unroll 2
    for (int qb = 0; qb < K1_; qb += 4) {
        int qa = qb + 2 * half;                // this lane's 2 K indices
        v2f a;
        a.x = im2col_load(bp, qa,     y, x);
        a.y = im2col_load(bp, qa + 1, y, x);

        int qrow = (qa >> 1) * 18;
        v2f bm0 = wp1[qrow + m];
        v2f bm1 = wp1[qrow + 16 + m];          // pad keeps read in-bounds
        bm1.x *= gmask; bm1.y *= gmask;        // mask, no divergence

        acc[0] = __builtin_amdgcn_wmma_f32_16x16x4_f32(
            false, a, false, bm0, (short)0, acc[0], false, false);
        acc[1] = __builtin_amdgcn_wmma_f32_16x16x4_f32(
            false, a, false, bm1, (short)0, acc[1], false, false);
    }

    // store: lane holds pixels x = xb + 8*half + r, out channel = N
    {
        float bv = bias[m];
        float* op = offs + (size_t)(b * 18 + m) * HW_ + (size_t)y * W_
                         + xb + 8 * half;
#pragma unroll
        for (int r = 0; r < 8; ++r) op[r] = acc[0][r] + bv;
    }
    if (m < 2) {
        int o = 16 + m;
        float bv = bias[o];
        float* op = offs + (size_t)(b * 18 + o) * HW_ + (size_t)y * W_
                         + xb + 8 * half;
#pragma unroll
        for (int r = 0; r < 8; ++r) op[r] = acc[1][r] + bv;
    }
}

// ---------------------------------------------------------------------------
// Stage 2: deformable conv via V_WMMA_F32_16X16X4_F32.
// Wave tile: M = 16 pixels, N = 64 out-channels (4 N-tiles), K = 64 by 4.
// Bilinear corners fetched as row-pair b64 loads; border clamping folded
// into remapped per-tap weights. Per-tap transposed weights pair-packed in
// LDS (16 KB) -> ds_load_b64 B fragments. K-loop unrolled x4 so up to 16
// b64 gathers are in flight before the first consume (LOADcnt allows 63).
// ---------------------------------------------------------------------------
__global__ void __launch_bounds__(256) deform_conv_wmma_kernel(
    const float* __restrict__ content, const float* __restrict__ offset,
    const float* __restrict__ dc_w, const float* __restrict__ dc_b,
    float* __restrict__ out)
{
    __shared__ v2f wt2[C_ * C_ / 2];  // [(c/2)*64 + o] = {w[o][c], w[o][c+1]}

    const int tid  = threadIdx.x;
    const int lane = tid & 31;
    const int wave = tid >> 5;
    const int m    = lane & 15;
    const int half = lane >> 4;

    int tileIdx = blockIdx.x * 8 + wave;      // 16384 tiles exact cover
    int b   = tileIdx >> 12;
    int rem = tileIdx & 4095;
    int y   = rem >> 4;
    int xb  = (rem & 15) << 4;
    int x   = xb + m;

    const float* cbase = content + (size_t)b * C_ * HW_;
    const float* obase = offset + (size_t)b * 18 * HW_ + (size_t)y * W_ + x;

    v8f acc[4];
#pragma unroll
    for (int t = 0; t < 4; ++t) acc[t] = (v8f){};

    for (int k = 0; k < 9; ++k) {
        // ---- stage pair-packed transposed tap weights into LDS ----
        __syncthreads();                       // previous tap's reads done
        for (int i = tid; i < C_ * C_ / 2; i += 256) {
            int c2 = i >> 6, o = i & 63;
            const float* src = dc_w + o * (C_ * 9) + (2 * c2) * 9 + k;
            v2f pv; pv.x = src[0]; pv.y = src[9];
            wt2[i] = pv;
        }
        __syncthreads();

        // ---- per-pixel bilinear setup (shared across all 64 channels) ----
        int ky = k / 3, kx = k % 3;
        float oy = obase[(size_t)(2 * k) * HW_];
        float ox = obase[(size_t)(2 * k + 1) * HW_];
        float py = (float)(y + ky - 1) + oy;
        float px = (float)(x + kx - 1) + ox;
        float y0f = floorf(py), x0f = floorf(px);
        float wy = py - y0f,    wx = px - x0f;
        int y0 = (int)y0f, x0i = (int)x0f;
        int y1 = y0 + 1;
        float vy0 = (y0 >= 0 && y0 < H_) ? 1.f : 0.f;
        float vy1 = (y1 >= 0 && y1 < H_) ? 1.f : 0.f;
        int y0c = min(max(y0, 0), H_ - 1), y1c = min(max(y1, 0), H_ - 1);

        // x corners are adjacent: one b64 per row. Remap weights so the
        // clamped-edge cases land on the right slot with weight 0 elsewhere.
        int  xs  = min(max(x0i, 0), W_ - 2);           // [0, 254]
        bool xin = (x0i >= 0) && (x0i < W_ - 1);
        float ax = xin ? (1.f - wx) : ((x0i == -1)     ? wx         : 0.f);
        float ay = xin ? wx         : ((x0i == W_ - 1) ? (1.f - wx) : 0.f);
        float ty = (1.f - wy) * vy0;
        float by = wy * vy1;
        float wtx = ty * ax, wty = ty * ay, wbx = by * ax, wby = by * ay;
        int rowt = y0c * W_ + xs;
        int rowb = y1c * W_ + xs;

        // ---- K loop: 2 channels/lane, b64 corner loads, 4 WMMAs ----
#pragma unroll 4
        for (int kb = 0; kb < C_; kb += 4) {
            int ca = kb + 2 * half;            // this lane's 2 channels
            const float* p0 = cbase + (size_t)ca * HW_;
            const float* p1 = p0 + HW_;
            v2f t0 = load2(p0 + rowt);
            v2f b0 = load2(p0 + rowb);
            v2f t1 = load2(p1 + rowt);
            v2f b1 = load2(p1 + rowb);
            v2f a;
            a.x = t0.x * wtx + t0.y * wty + b0.x * wbx + b0.y * wby;
            a.y = t1.x * wtx + t1.y * wty + b1.x * wbx + b1.y * wby;

            int crow = ((ca >> 1) << 6);       // (ca/2)*64
            v2f bm0 = wt2[crow + m];
            v2f bm1 = wt2[crow + 16 + m];
            v2f bm2 = wt2[crow + 32 + m];
            v2f bm3 = wt2[crow + 48 + m];

            acc[0] = __builtin_amdgcn_wmma_f32_16x16x4_f32(
                false, a, false, bm0, (short)0, acc[0], false, false);
            acc[1] = __builtin_amdgcn_wmma_f32_16x16x4_f32(
                false, a, false, bm1, (short)0, acc[1], false, false);
            acc[2] = __builtin_amdgcn_wmma_f32_16x16x4_f32(
                false, a, false, bm2, (short)0, acc[2], false, false);
            acc[3] = __builtin_amdgcn_wmma_f32_16x16x4_f32(
                false, a, false, bm3, (short)0, acc[3], false, false);
        }
    }

    // ---- store D + bias: lane holds pixels x = xb + 8*half + r, N = m ----
#pragma unroll
    for (int t = 0; t < 4; ++t) {
        int o = t * 16 + m;
        float bv = dc_b[o];
        float* op = out + (size_t)(b * C_ + o) * HW_ + (size_t)y * W_
                        + xb + 8 * half;
#pragma unroll
        for (int r = 0; r < 8; ++r) op[r] = acc[t][r] + bv;
    }
}

// ---------------------------------------------------------------------------
extern "C" void kernel_launch(void* const* d_in, const int* in_sizes, int n_in,
                              void* d_out, int out_size, void* d_ws, size_t ws_size,
                              hipStream_t stream) {
    const float* content  = (const float*)d_in[0];
    const float* blur     = (const float*)d_in[1];
    const float* offset_w = (const float*)d_in[2];
    const float* offset_b = (const float*)d_in[3];
    const float* dc_w     = (const float*)d_in[4];
    const float* dc_b     = (const float*)d_in[5];

    float* out    = (float*)d_out;            // (4,64,256,256)
    float* offset = out + OUT0_ELEMS;         // (4,18,256,256), second tuple elem

    // Stage 1: WMMA im2col offset conv (writes d_out's second region)
    offset_conv_wmma_kernel<<<2048, 256, 0, stream>>>(
        blur, offset_w, offset_b, offset);

    // Stage 2: WMMA deformable conv (consumes offsets via stream ordering)
    deform_conv_wmma_kernel<<<2048, 256, 0, stream>>>(
        content, offset, dc_w, dc_b, out);
}